// MultiHeadAttentionLayer_42820823941538
// MI455X (gfx1250) — compile-verified
//
#include <hip/hip_runtime.h>
#include <hip/hip_bf16.h>
#include <math.h>

// ---------------------------------------------------------------------------
// Graph multi-head attention (N=100000, E=1.6M, IN=128, H=8, D=8, R=3)
//
// Phase 1: QKV projections via V_WMMA_F32_16X16X4_F32 (full fp32 precision).
// Phase 2: per-edge score/exp + scatter-add with global_atomic_add_f32
//          (accumulators are L2-resident on MI455X's 192MB L2).
// Phase 3: normalize by z.
// ---------------------------------------------------------------------------

typedef __attribute__((ext_vector_type(2))) float v2f;
typedef __attribute__((ext_vector_type(8))) float v8f;

#define IN_DIM   128
#define HD       64      // H*D
#define NHEADS   8
#define DHEAD    8
#define INV_SCALE 0.3535533905932738f   // 1/sqrt(8)

// ---------------------------------------------------------------------------
// Phase 1: Y = h @ W^T + b for W in {Wq,Wk,Wv}, each [64,128] row-major.
// One wave computes one 16x16 output tile; 12 waves per block cover the
// 3*64 = 192 output columns for one 16-row tile.
//
// fp32 WMMA frag layouts (ISA 7.12.2):
//   A 16x4:  lane L holds row m=L&15; VGPR0/1 = A[m][kk+2*(L>>4) + {0,1}]
//   B 4x16:  lane L holds col n=L&15; VGPR0/1 = B[kk+2*(L>>4)+{0,1}][n]
//            (B[k][n] == W[n][k] since Y = h @ W^T)
//   C/D 16x16: VGPR r, lane L -> element (m = r + 8*(L>>4), n = L&15)
// Both A and B frags are a contiguous float2 per lane -> global_load_b64.
// ---------------------------------------------------------------------------
__global__ __launch_bounds__(384) void qkv_wmma_kernel(
    const float* __restrict__ h,
    const float* __restrict__ Wq, const float* __restrict__ bq,
    const float* __restrict__ Wk, const float* __restrict__ bk,
    const float* __restrict__ Wv, const float* __restrict__ bv,
    float* __restrict__ Qt, float* __restrict__ Kt, float* __restrict__ Vt)
{
    const int wave   = threadIdx.x >> 5;     // 0..11: col tile
    const int lane   = threadIdx.x & 31;
    const int row0   = blockIdx.x << 4;      // 16 rows per block
    const int mat    = wave >> 2;            // 0=Q, 1=K, 2=V
    const int colLoc = (wave & 3) << 4;      // 0,16,32,48 inside [64]

    const float* __restrict__ W = (mat == 0) ? Wq : (mat == 1) ? Wk : Wv;
    const float* __restrict__ b = (mat == 0) ? bq : (mat == 1) ? bk : bv;
    float* __restrict__ Y       = (mat == 0) ? Qt : (mat == 1) ? Kt : Vt;

    const int laneLo = lane & 15;
    const int laneHi = lane >> 4;

    const float* aPtr = h + (size_t)(row0   + laneLo) * IN_DIM + 2 * laneHi;
    const float* bPtr = W + (size_t)(colLoc + laneLo) * IN_DIM + 2 * laneHi;

    v8f c = {0.f, 0.f, 0.f, 0.f, 0.f, 0.f, 0.f, 0.f};

    #pragma unroll 4
    for (int kk = 0; kk < IN_DIM; kk += 4) {
        v2f a  = *(const v2f*)(aPtr + kk);
        v2f bb = *(const v2f*)(bPtr + kk);
        // D = A(16x4,f32) x B(4x16,f32) + C   (8 args, fp32 WMMA)
        c = __builtin_amdgcn_wmma_f32_16x16x4_f32(
                /*neg_a=*/false, a, /*neg_b=*/false, bb,
                /*c_mod=*/(short)0, c, /*reuse_a=*/false, /*reuse_b=*/false);
    }

    const float bias = b[colLoc + laneLo];
    #pragma unroll
    for (int r = 0; r < 8; ++r) {
        const int row = row0 + r + 8 * laneHi;
        Y[(size_t)row * HD + colLoc + laneLo] = c[r] + bias;
    }
}

// ---------------------------------------------------------------------------
// Phase 2: one lane per (edge, head). 8 lanes of an edge gather a contiguous
// 256B span of K[src]/Q[dst]/V[src] (coalesced); scatter via fp32 L2 atomics.
// ---------------------------------------------------------------------------
__global__ __launch_bounds__(256) void edge_attn_kernel(
    const float* __restrict__ Qt, const float* __restrict__ Kt,
    const float* __restrict__ Vt,
    const float* __restrict__ pe, const float* __restrict__ Ww,
    const float* __restrict__ bw,
    const int* __restrict__ src, const int* __restrict__ dst,
    float* __restrict__ outWV, float* __restrict__ z, int E)
{
    const long long gid = (long long)blockIdx.x * blockDim.x + threadIdx.x;
    const int e  = (int)(gid >> 3);
    const int hh = (int)(gid & 7);
    if (e >= E) return;

    const int s = src[e];
    const int d = dst[e];

    const float4* kp = (const float4*)(Kt + (size_t)s * HD + hh * DHEAD);
    const float4* qp = (const float4*)(Qt + (size_t)d * HD + hh * DHEAD);
    const float4 k0 = kp[0], k1 = kp[1];
    const float4 q0 = qp[0], q1 = qp[1];

    float score = k0.x * q0.x + k0.y * q0.y + k0.z * q0.z + k0.w * q0.w
                + k1.x * q1.x + k1.y * q1.y + k1.z * q1.z + k1.w * q1.w;
    score *= INV_SCALE;

    // att bias shared across heads: pe[e] . Ww + bw
    const float bias = pe[(size_t)e * 3 + 0] * Ww[0]
                     + pe[(size_t)e * 3 + 1] * Ww[1]
                     + pe[(size_t)e * 3 + 2] * Ww[2] + bw[0];

    float x = score + bias;
    x = fminf(5.0f, fmaxf(-5.0f, x));
    const float w = __expf(x);

    const float4* vp = (const float4*)(Vt + (size_t)s * HD + hh * DHEAD);
    const float4 v0 = vp[0], v1 = vp[1];

    float* o = outWV + (size_t)d * HD + hh * DHEAD;
    __hip_atomic_fetch_add(o + 0, w * v0.x, __ATOMIC_RELAXED, __HIP_MEMORY_SCOPE_AGENT);
    __hip_atomic_fetch_add(o + 1, w * v0.y, __ATOMIC_RELAXED, __HIP_MEMORY_SCOPE_AGENT);
    __hip_atomic_fetch_add(o + 2, w * v0.z, __ATOMIC_RELAXED, __HIP_MEMORY_SCOPE_AGENT);
    __hip_atomic_fetch_add(o + 3, w * v0.w, __ATOMIC_RELAXED, __HIP_MEMORY_SCOPE_AGENT);
    __hip_atomic_fetch_add(o + 4, w * v1.x, __ATOMIC_RELAXED, __HIP_MEMORY_SCOPE_AGENT);
    __hip_atomic_fetch_add(o + 5, w * v1.y, __ATOMIC_RELAXED, __HIP_MEMORY_SCOPE_AGENT);
    __hip_atomic_fetch_add(o + 6, w * v1.z, __ATOMIC_RELAXED, __HIP_MEMORY_SCOPE_AGENT);
    __hip_atomic_fetch_add(o + 7, w * v1.w, __ATOMIC_RELAXED, __HIP_MEMORY_SCOPE_AGENT);
    __hip_atomic_fetch_add(z + (size_t)d * NHEADS + hh, w,
                           __ATOMIC_RELAXED, __HIP_MEMORY_SCOPE_AGENT);
}

// ---------------------------------------------------------------------------
// Phase 3: out /= (z + 1e-6), one thread per output element.
// ---------------------------------------------------------------------------
__global__ __launch_bounds__(256) void normalize_kernel(
    float* __restrict__ out, const float* __restrict__ z, int total)
{
    const int i = blockIdx.x * blockDim.x + threadIdx.x;
    if (i >= total) return;
    const int n  = i >> 6;          // node
    const int hh = (i >> 3) & 7;    // head
    out[i] = out[i] / (z[(size_t)n * NHEADS + hh] + 1e-6f);
}

extern "C" void kernel_launch(void* const* d_in, const int* in_sizes, int n_in,
                              void* d_out, int out_size, void* d_ws, size_t ws_size,
                              hipStream_t stream) {
    const float* h  = (const float*)d_in[0];
    const float* pe = (const float*)d_in[1];
    const float* Wq = (const float*)d_in[2];
    const float* bq = (const float*)d_in[3];
    const float* Wk = (const float*)d_in[4];
    const float* bk = (const float*)d_in[5];
    const float* Wv = (const float*)d_in[6];
    const float* bv = (const float*)d_in[7];
    const float* Ww = (const float*)d_in[8];
    const float* bw = (const float*)d_in[9];
    const int*  src = (const int*)d_in[10];
    const int*  dst = (const int*)d_in[11];

    const int N = in_sizes[0] / IN_DIM;   // 100000 (divisible by 16)
    const int E = in_sizes[10];           // 1600000

    float* out = (float*)d_out;           // [N, H, D] = [N, 64], used as wV accum
    float* Qt  = (float*)d_ws;            // [N, 64]
    float* Kt  = Qt + (size_t)N * HD;     // [N, 64]
    float* Vt  = Kt + (size_t)N * HD;     // [N, 64]
    float* z   = Vt + (size_t)N * HD;     // [N, 8]

    // zero accumulators (graph-capture-safe async memsets)
    hipMemsetAsync(out, 0, (size_t)N * HD * sizeof(float), stream);
    hipMemsetAsync(z,   0, (size_t)N * NHEADS * sizeof(float), stream);

    // Phase 1: QKV projections (WMMA). 12 waves/block = 12 col tiles.
    qkv_wmma_kernel<<<N / 16, 384, 0, stream>>>(h, Wq, bq, Wk, bk, Wv, bv,
                                                Qt, Kt, Vt);

    // Phase 2: edge scatter. 8 lanes per edge.
    {
        const long long threads = (long long)E * NHEADS;
        const int blocks = (int)((threads + 255) / 256);
        edge_attn_kernel<<<blocks, 256, 0, stream>>>(Qt, Kt, Vt, pe, Ww, bw,
                                                     src, dst, out, z, E);
    }

    // Phase 3: normalize.
    {
        const int total = N * HD;
        normalize_kernel<<<(total + 255) / 256, 256, 0, stream>>>(out, z, total);
    }
}